// SAGPool_48112223650039
// MI455X (gfx1250) — compile-verified
//
#include <hip/hip_runtime.h>
#include <hip/hip_bf16.h>

// ---------------- static problem configuration (matches reference) ----------
constexpr int G_   = 1024;       // graphs
constexpr int NPG_ = 512;        // nodes per graph
constexpr int N_   = G_ * NPG_;  // 524288 nodes
constexpr int F_   = 256;        // feature dim
constexpr int E_   = 8388608;    // edges
constexpr int K_   = 256;        // kept nodes per graph

typedef float v2f __attribute__((ext_vector_type(2)));
typedef float v8f __attribute__((ext_vector_type(8)));

// ---------------------------------------------------------------------------
// Kernel 1: h = x @ w  via V_WMMA_F32_16X16X4_F32.
// One wave32 handles 16 nodes. B = w broadcast across all 16 columns, so every
// column of C equals h[16 nodes]. A-layout (16x4 f32): lane L -> M = L&15,
// K = {0,1} + 2*(L>>4)  => one contiguous float2 per lane per K-step of 4.
// C-layout (16x16 f32): VGPR v, lanes 0-15 -> M=v, lanes 16-31 -> M=v+8.
// Also fuses zero-init of agg[] for this block's node range.
// ---------------------------------------------------------------------------
__global__ __launch_bounds__(256) void gemv_wmma_kernel(
    const float* __restrict__ x, const float* __restrict__ w,
    float* __restrict__ h, float* __restrict__ agg) {
  __shared__ float wl[F_];
  const int tid = threadIdx.x;
  wl[tid] = w[tid];                         // blockDim.x == 256 == F_
  const int nodeBase = blockIdx.x * 128;    // 8 waves * 16 nodes
  if (tid < 128) agg[nodeBase + tid] = 0.0f;
  __syncthreads();

  const int wave  = tid >> 5;
  const int lane  = tid & 31;
  const int m     = lane & 15;
  const int khalf = lane >> 4;              // 0 or 1
  const int rowBase = nodeBase + wave * 16;
  const float* __restrict__ xrow =
      x + (size_t)(rowBase + m) * F_ + 2 * khalf;
  const float* __restrict__ wptr = &wl[2 * khalf];

  v8f acc = {0.f, 0.f, 0.f, 0.f, 0.f, 0.f, 0.f, 0.f};
#pragma unroll 4
  for (int k = 0; k < F_; k += 4) {
    v2f a = *(const v2f*)(xrow + k);        // x[row][k+2*khalf .. +1]
    v2f b = *(const v2f*)(wptr + k);        // w[k+2*khalf .. +1]
    acc = __builtin_amdgcn_wmma_f32_16x16x4_f32(
        false, a, false, b, (short)0, acc, false, false);
  }
  // Columns are identical; lanes with N==0 (lanes 0 and 16) hold all 16 M's.
  if (m == 0) {
    const int mb = rowBase + 8 * khalf;
#pragma unroll
    for (int j = 0; j < 8; ++j) h[mb + j] = acc[j];
  }
}

// ---------------------------------------------------------------------------
// Kernel 2: agg[dst] += ew * h[src]   (native f32 global atomics, L2-resident)
// ---------------------------------------------------------------------------
__global__ __launch_bounds__(256) void edge_scatter_kernel(
    const int* __restrict__ src, const int* __restrict__ dst,
    const float* __restrict__ ew, const float* __restrict__ h,
    float* __restrict__ agg) {
  const int e = blockIdx.x * 256 + threadIdx.x;
  if (e < E_) {
    const float v = ew[e] * h[src[e]];
    unsafeAtomicAdd(&agg[dst[e]], v);
  }
}

// ---------------------------------------------------------------------------
// Kernel 3: per-graph score = tanh(agg+h); top-K via LDS bitonic sort on the
// total order (value desc, index asc) == jax.lax.top_k semantics. Writes
// top_score, kept node ids, new_id relabel table, and graph-index output.
// ---------------------------------------------------------------------------
__global__ __launch_bounds__(512) void topk_kernel(
    const float* __restrict__ h, const float* __restrict__ agg,
    int* __restrict__ new_id, float* __restrict__ top_score,
    int* __restrict__ kept, int* __restrict__ out_ngi) {
  __shared__ float sval[NPG_];
  __shared__ int   sidx[NPG_];
  const int g = blockIdx.x;
  const int t = threadIdx.x;
  const int gnode = g * NPG_ + t;
  sval[t] = tanhf(agg[gnode] + h[gnode]);
  sidx[t] = t;
  new_id[gnode] = -1;
  __syncthreads();

  for (int k = 2; k <= NPG_; k <<= 1) {
    for (int j = k >> 1; j > 0; j >>= 1) {
      const int ixj = t ^ j;
      if (ixj > t) {
        const float v1 = sval[t], v2 = sval[ixj];
        const int   i1 = sidx[t], i2 = sidx[ixj];
        // "before" order: higher value first; ties -> lower index first
        const bool before21 = (v2 > v1) || (v2 == v1 && i2 < i1);
        const bool forward  = ((t & k) == 0);
        if (forward == before21) {
          sval[t] = v2; sval[ixj] = v1;
          sidx[t] = i2; sidx[ixj] = i1;
        }
      }
      __syncthreads();
    }
  }

  if (t < K_) {
    const float s     = sval[t];
    const int   local = sidx[t];
    const int   node  = g * NPG_ + local;
    const int   pos   = g * K_ + t;
    top_score[pos] = s;
    kept[pos]      = node;
    out_ngi[pos]   = g;
    new_id[node]   = pos;   // -1 init ordered before via barriers in the sort
  }
}

// ---------------------------------------------------------------------------
// Kernel 4: pooled_x[i] = x[kept[i]] * top_score[i]   (float4 vectorized)
// ---------------------------------------------------------------------------
__global__ __launch_bounds__(256) void gather_scale_kernel(
    const float* __restrict__ x, const int* __restrict__ kept,
    const float* __restrict__ top_score, float* __restrict__ out_x) {
  const long gid = (long)blockIdx.x * 256 + threadIdx.x; // one float4 each
  const int row = (int)(gid >> 6);                       // F_/4 = 64 per row
  const int c   = (int)(gid & 63);
  const int node = kept[row];
  const float s  = top_score[row];
  float4 v = ((const float4*)(x + (size_t)node * F_))[c];
  v.x *= s; v.y *= s; v.z *= s; v.w *= s;
  ((float4*)(out_x + (size_t)row * F_))[c] = v;
}

// ---------------------------------------------------------------------------
// Kernel 5: edge relabel + validity masking
// ---------------------------------------------------------------------------
__global__ __launch_bounds__(256) void edge_reindex_kernel(
    const int* __restrict__ src, const int* __restrict__ dst,
    const float* __restrict__ ew, const int* __restrict__ new_id,
    int* __restrict__ out_src, int* __restrict__ out_dst,
    float* __restrict__ out_ew) {
  const int e = blockIdx.x * 256 + threadIdx.x;
  if (e < E_) {
    const int ns = new_id[src[e]];
    const int nd = new_id[dst[e]];
    const bool valid = (ns >= 0) && (nd >= 0);
    out_src[e] = valid ? ns : 0;
    out_dst[e] = valid ? nd : 0;
    out_ew[e]  = valid ? ew[e] : 0.0f;
  }
}

// ---------------------------------------------------------------------------
extern "C" void kernel_launch(void* const* d_in, const int* in_sizes, int n_in,
                              void* d_out, int out_size, void* d_ws,
                              size_t ws_size, hipStream_t stream) {
  (void)in_sizes; (void)n_in; (void)out_size; (void)ws_size;
  const float* x  = (const float*)d_in[0];
  const int*   ei = (const int*)d_in[1];
  const float* ew = (const float*)d_in[2];
  // d_in[3] (node_graph_index) not needed: kept nodes of graph g map to g.
  const float* w  = (const float*)d_in[4];

  const int* src = ei;
  const int* dst = ei + E_;

  // workspace partition (~8 MB)
  char* ws = (char*)d_ws;
  float* h      = (float*)(ws);
  float* agg    = (float*)(ws + (size_t)N_ * 4);
  int*   new_id = (int*)  (ws + (size_t)N_ * 8);
  float* tops   = (float*)(ws + (size_t)N_ * 12);
  int*   kept   = (int*)  (ws + (size_t)N_ * 12 + (size_t)G_ * K_ * 4);

  // output partition (return order: pooled_x, edge_index, edge_weight, ngi)
  float* out_x   = (float*)d_out;
  int*   out_src = (int*)(out_x + (size_t)G_ * K_ * F_);
  int*   out_dst = out_src + E_;
  float* out_ew  = (float*)(out_dst + E_);
  int*   out_ngi = (int*)(out_ew + E_);

  gemv_wmma_kernel<<<N_ / 128, 256, 0, stream>>>(x, w, h, agg);
  edge_scatter_kernel<<<(E_ + 255) / 256, 256, 0, stream>>>(src, dst, ew, h, agg);
  topk_kernel<<<G_, 512, 0, stream>>>(h, agg, new_id, tops, kept, out_ngi);
  gather_scale_kernel<<<(G_ * K_ * (F_ / 4)) / 256, 256, 0, stream>>>(
      x, kept, tops, out_x);
  edge_reindex_kernel<<<(E_ + 255) / 256, 256, 0, stream>>>(
      src, dst, ew, new_id, out_src, out_dst, out_ew);
}